// FairGCNAutoEncoder_15290083573912
// MI455X (gfx1250) — compile-verified
//
#include <hip/hip_runtime.h>
#include <hip/hip_bf16.h>

typedef float v2f __attribute__((ext_vector_type(2)));
typedef float v8f __attribute__((ext_vector_type(8)));

#define N_NODES 50000
#define N_EDGES 800000
#define CH      128
#define OUT_ELEMS (N_NODES * CH)   // 6,400,000
#define M_TILES (N_NODES / 16)     // 3125 (exact)

// ---------------------------------------------------------------------------
// 1) Zero the output accumulator and degree array (must run every launch:
//    harness poisons once, graph replays many times).
// ---------------------------------------------------------------------------
__global__ __launch_bounds__(256) void init_zero(float* __restrict__ out,
                                                 float* __restrict__ deg) {
    int tid = blockIdx.x * 256 + threadIdx.x;
    if (tid < OUT_ELEMS) out[tid] = 0.0f;
    if (tid < N_NODES)   deg[tid] = 0.0f;
}

// ---------------------------------------------------------------------------
// 2) Degree of dst (self-loop +1 folded in later).
// ---------------------------------------------------------------------------
__global__ __launch_bounds__(256) void degree_count(const int* __restrict__ ei,
                                                    float* __restrict__ deg) {
    int e = blockIdx.x * 256 + threadIdx.x;
    if (e < N_EDGES) {
        int d = ei[N_EDGES + e];           // edge_index[1][e]
        atomicAdd(&deg[d], 1.0f);
    }
}

// ---------------------------------------------------------------------------
// 3) dinv = rsqrt(deg + 1)   (deg >= 1 always, thanks to self loop)
// ---------------------------------------------------------------------------
__global__ __launch_bounds__(256) void deg_to_dinv(float* __restrict__ deg) {
    int i = blockIdx.x * 256 + threadIdx.x;
    if (i < N_NODES) deg[i] = rsqrtf(deg[i] + 1.0f);
}

// ---------------------------------------------------------------------------
// 4) H = X @ W via V_WMMA_F32_16X16X4_F32 (fp32 matrix pipe, wave32).
//    W (64 KB) staged once per block into LDS, SWIZZLED as k-pairs:
//      sW2[p][n] = { W[2p][n], W[2p+1][n] }   (p = 0..63, n = 0..127)
//    so every WMMA B operand is one even-aligned ds_load_b64 with an
//    immediate offset -- no register re-assembly moves in the hot loop.
//    One wave -> 16 rows x 128 cols (8 v8f accumulators).
//    A 16x4 f32 layout: lanes 0-15 = M rows; VGPR0 = K+0/K+2 (lane[4] picks
//    the +2 half), VGPR1 = K+1/K+3. B mirrored. D per 16x16 f32 C/D table.
// ---------------------------------------------------------------------------
__global__ __launch_bounds__(256) void gemm_wmma(const float* __restrict__ X,
                                                 const float* __restrict__ W,
                                                 float* __restrict__ H) {
    __shared__ float sW[CH * CH];          // 64 KB, pair-swizzled view

    // Cooperative swizzled stage: i -> (pair p, col n). Reads are two
    // coalesced row streams; writes are contiguous float2.
    for (int i = threadIdx.x; i < (CH / 2) * CH; i += 256) {
        int p = i >> 7;        // k-pair index 0..63
        int n = i & (CH - 1);  // column 0..127
        float2 w2;
        w2.x = W[(size_t)(2 * p)     * CH + n];
        w2.y = W[(size_t)(2 * p + 1) * CH + n];
        reinterpret_cast<float2*>(sW)[i] = w2;
    }
    __syncthreads();

    const int wave = blockIdx.x * 8 + (threadIdx.x >> 5);   // uniform per wave
    if (wave < M_TILES) {                                   // whole-wave guard
        const int lane = threadIdx.x & 31;
        const int m    = lane & 15;
        const int khh  = lane >> 4;          // 0 or 1: which k-pair half
        const int kh   = khh << 1;           // 0 or 2
        const int row0 = wave << 4;

        const float* xrow = X + (size_t)(row0 + m) * CH;
        // per-lane base into pair-swizzled W: pair (khh), column m
        const float* bw = sW + (size_t)(khh * CH + m) * 2;

        v8f acc[8] = {};

#pragma unroll 4
        for (int k = 0; k < CH; k += 4) {
            v2f a;
            a[0] = xrow[k + kh];
            a[1] = xrow[k + kh + 1];
            const float* bk = bw + (size_t)(k >> 1) * (CH * 2);
#pragma unroll
            for (int t = 0; t < 8; ++t) {
                // one aligned ds_load_b64: {W[k+kh][n], W[k+kh+1][n]}
                v2f b = *reinterpret_cast<const v2f*>(bk + (t << 5));
                acc[t] = __builtin_amdgcn_wmma_f32_16x16x4_f32(
                    /*neg_a=*/false, a, /*neg_b=*/false, b,
                    /*c_mod=*/(short)0, acc[t],
                    /*reuse_a=*/false, /*reuse_b=*/false);
            }
        }

        // D layout: VGPR v -> row (v + 8*lane[4]), col = lane&15 in tile t.
        const int mh  = khh << 3;            // 0 or 8
        const int nlo = lane & 15;
#pragma unroll
        for (int t = 0; t < 8; ++t) {
#pragma unroll
            for (int v = 0; v < 8; ++v) {
                H[(size_t)(row0 + mh + v) * CH + (t << 4) + nlo] = acc[t][v];
            }
        }
    }
}

// ---------------------------------------------------------------------------
// 5) Edge scatter: one wave per edge, 4 channels per lane (float4 gather from
//    L2-resident H, 4x global_atomic_add_f32 into out). Edge id is forced
//    into an SGPR via readfirstlane so src/dst/dinv fetches become scalar
//    loads instead of 32-lane broadcast vector loads.
// ---------------------------------------------------------------------------
__global__ __launch_bounds__(256) void scatter_edges(const int* __restrict__ ei,
                                                     const float* __restrict__ H,
                                                     const float* __restrict__ dinv,
                                                     float* __restrict__ out) {
    const int e = __builtin_amdgcn_readfirstlane(blockIdx.x * 8 + (threadIdx.x >> 5));
    if (e >= N_EDGES) return;
    const int lane = threadIdx.x & 31;
    const int s = ei[e];
    const int d = ei[N_EDGES + e];
    const float norm = dinv[s] * dinv[d];

    const float4 v = *reinterpret_cast<const float4*>(H + (size_t)s * CH + lane * 4);
    float* o = out + (size_t)d * CH + lane * 4;
    atomicAdd(o + 0, v.x * norm);
    atomicAdd(o + 1, v.y * norm);
    atomicAdd(o + 2, v.z * norm);
    atomicAdd(o + 3, v.w * norm);
}

// ---------------------------------------------------------------------------
// 6) Self-loop term (dinv[i]^2 * h[i]) + bias + ReLU, in place on out.
// ---------------------------------------------------------------------------
__global__ __launch_bounds__(256) void finalize(float* __restrict__ out,
                                                const float* __restrict__ H,
                                                const float* __restrict__ dinv,
                                                const float* __restrict__ bias) {
    int tid = blockIdx.x * 256 + threadIdx.x;
    if (tid >= OUT_ELEMS) return;
    int i = tid >> 7;
    int c = tid & (CH - 1);
    float dn = dinv[i];
    float v = out[tid] + dn * dn * H[tid] + bias[c];
    out[tid] = fmaxf(v, 0.0f);
}

// ---------------------------------------------------------------------------
extern "C" void kernel_launch(void* const* d_in, const int* in_sizes, int n_in,
                              void* d_out, int out_size, void* d_ws, size_t ws_size,
                              hipStream_t stream) {
    const float* x  = (const float*)d_in[0];   // [50000,128]
    const int*   ei = (const int*)d_in[1];     // [2,800000] flat
    const float* W  = (const float*)d_in[2];   // [128,128]
    const float* b  = (const float*)d_in[3];   // [128]
    float* out = (float*)d_out;                // [50000,128]

    float* deg = (float*)d_ws;                 // 50000 floats (becomes dinv)
    float* H   = (float*)d_ws + 51200;         // 6,400,000 floats

    init_zero    <<<(OUT_ELEMS + 255) / 256, 256, 0, stream>>>(out, deg);
    degree_count <<<(N_EDGES   + 255) / 256, 256, 0, stream>>>(ei, deg);
    deg_to_dinv  <<<(N_NODES   + 255) / 256, 256, 0, stream>>>(deg);
    gemm_wmma    <<<(M_TILES + 7) / 8, 256, 0, stream>>>(x, W, H);
    scatter_edges<<<N_EDGES / 8, 256, 0, stream>>>(ei, H, deg, out);
    finalize     <<<(OUT_ELEMS + 255) / 256, 256, 0, stream>>>(out, H, deg, b);
}